// Point_Embedding_13718125543638
// MI455X (gfx1250) — compile-verified
//
#include <hip/hip_runtime.h>
#include <hip/hip_bf16.h>
#include <math.h>

typedef __attribute__((ext_vector_type(2))) float v2f;
typedef __attribute__((ext_vector_type(8))) float v8f;

#define B_       8
#define N_       4096
#define INIT_D_  6
#define DIMS_    64
#define KNB_     32
#define R2_      (0.1f * 0.1f)
#define EPS_     1e-5f

// ---------------------------------------------------------------------------
// Kernel 1: x = relu(layer_norm(inputs @ W1 + b1, g1, be1))
// grid = B*N blocks, block = 64 threads (2 waves), one channel per thread.
// ---------------------------------------------------------------------------
__global__ void embed_kernel(const float* __restrict__ inp,
                             const float* __restrict__ W1,
                             const float* __restrict__ b1,
                             const float* __restrict__ g1,
                             const float* __restrict__ be1,
                             float* __restrict__ x) {
  const int row = blockIdx.x;        // 0 .. B*N-1
  const int ch  = threadIdx.x;       // 0 .. 63
  const float* ir = inp + row * INIT_D_;

  float acc = b1[ch];
#pragma unroll
  for (int i = 0; i < INIT_D_; ++i)
    acc = fmaf(ir[i], W1[i * DIMS_ + ch], acc);

  __shared__ float red[2][2];
  const int wid = ch >> 5;

  // mean
  float s = acc;
#pragma unroll
  for (int o = 16; o > 0; o >>= 1) s += __shfl_xor(s, o, 32);
  if ((ch & 31) == 0) red[wid][0] = s;
  __syncthreads();
  const float m = (red[0][0] + red[1][0]) * (1.0f / DIMS_);

  // variance (two-pass, matches reference)
  const float d = acc - m;
  float s2 = d * d;
#pragma unroll
  for (int o = 16; o > 0; o >>= 1) s2 += __shfl_xor(s2, o, 32);
  if ((ch & 31) == 0) red[wid][1] = s2;
  __syncthreads();
  const float v = (red[0][1] + red[1][1]) * (1.0f / DIMS_);

  const float val = d * rsqrtf(v + EPS_) * g1[ch] + be1[ch];
  x[row * DIMS_ + ch] = fmaxf(val, 0.0f);
}

// ---------------------------------------------------------------------------
// Kernel 2: query_ball — first 32 indices (ascending) with dist <= r^2.
// One wave32 per row; float4 streaming (512B/wave/iter) over the 512MB matrix
// (the whole problem's HBM roofline: ~22us @ 23.3 TB/s). Ordered extraction
// via ballot + popcount prefix ranks; uniform early exit at 32 hits.
// ---------------------------------------------------------------------------
__global__ void qball_kernel(const float* __restrict__ dmat,
                             int* __restrict__ idx) {
  const int lane = threadIdx.x & 31;
  const int wid  = threadIdx.x >> 5;
  const int row  = blockIdx.x * 8 + wid;     // 0 .. B*N-1
  const float4* drow = (const float4*)(dmat + (size_t)row * N_);
  int* orow = idx + row * KNB_;

  int collected = 0;
  int first = -1;
  const unsigned mlt = (1u << lane) - 1u;

  for (int it = 0; it < N_ / 128; ++it) {
    const float4 dq = drow[it * 32 + lane];
    const unsigned m0 = __builtin_amdgcn_ballot_w32(dq.x <= R2_);
    const unsigned m1 = __builtin_amdgcn_ballot_w32(dq.y <= R2_);
    const unsigned m2 = __builtin_amdgcn_ballot_w32(dq.z <= R2_);
    const unsigned m3 = __builtin_amdgcn_ballot_w32(dq.w <= R2_);
    const int total = __popc(m0) + __popc(m1) + __popc(m2) + __popc(m3);
    if (total) {
      const int base = it * 128;
      if (first < 0) {
        int best = 0x7fffffff;
        if (m0) best = min(best, 4 * (int)__builtin_ctz(m0) + 0);
        if (m1) best = min(best, 4 * (int)__builtin_ctz(m1) + 1);
        if (m2) best = min(best, 4 * (int)__builtin_ctz(m2) + 2);
        if (m3) best = min(best, 4 * (int)__builtin_ctz(m3) + 3);
        first = base + best;
      }
      // rank of this lane's hits among the iteration, in element order
      // (element = base + lane*4 + c; order is lane-major)
      int r = __popc(m0 & mlt) + __popc(m1 & mlt) +
              __popc(m2 & mlt) + __popc(m3 & mlt);
      if ((m0 >> lane) & 1u) { if (collected + r < KNB_) orow[collected + r] = base + lane * 4 + 0; ++r; }
      if ((m1 >> lane) & 1u) { if (collected + r < KNB_) orow[collected + r] = base + lane * 4 + 1; ++r; }
      if ((m2 >> lane) & 1u) { if (collected + r < KNB_) orow[collected + r] = base + lane * 4 + 2; ++r; }
      if ((m3 >> lane) & 1u) { if (collected + r < KNB_) orow[collected + r] = base + lane * 4 + 3; ++r; }
      collected += total;
      if (collected >= KNB_) break;      // wave-uniform
    }
  }
  if (collected < KNB_) {                 // pad with first hit (reference semantics)
    const int p = collected + lane;
    if (p < KNB_) orow[p] = first;
  }
}

// ---------------------------------------------------------------------------
// Kernel 3: PQ = x @ [ W2a | (W2b - W2a) ]   (32768x64 @ 64x128, fp32 WMMA)
// Block = 256 threads = 8 waves: one 16-row tile, each wave one 16-col tile,
// 16 chained V_WMMA_F32_16X16X4_F32 over K=64. x tile staged in LDS with
// stride 68 (bank-conflict-free for the column-strided A-fragment reads).
// ---------------------------------------------------------------------------
#define XS_STRIDE 68
__global__ void project_kernel(const float* __restrict__ x,
                               const float* __restrict__ W2,
                               float* __restrict__ PQ) {
  __shared__ float xs[16 * XS_STRIDE];
  __shared__ float wc[64 * 128];
  const int tid = threadIdx.x;             // 0..255
  const int rowbase = blockIdx.x * 16;

  // stage x tile (16 x 64), coalesced
  for (int p = tid; p < 16 * 64; p += 256) {
    const int r = p >> 6, c = p & 63;
    xs[r * XS_STRIDE + c] = x[(size_t)(rowbase + r) * DIMS_ + c];
  }
  // stage combined weights: wc[k][n<64] = W2a[k][n]; wc[k][64+n] = W2b[k][n]-W2a[k][n]
  for (int p = tid; p < 64 * 128; p += 256) {
    const int k = p >> 7, n = p & 127;
    float w;
    if (n < 64) w = W2[k * 64 + n];
    else        w = W2[(64 + k) * 64 + (n - 64)] - W2[k * 64 + (n - 64)];
    wc[p] = w;
  }
  __syncthreads();

  const int lane = tid & 31;
  const int wv   = tid >> 5;               // 0..7 : column tile
  const int n0   = wv * 16;
  const int mn   = lane & 15;              // M for A, N for B/D
  const int kb   = (lane >> 4) * 2;        // half-wave K pair select

  v8f c = {0.f, 0.f, 0.f, 0.f, 0.f, 0.f, 0.f, 0.f};
#pragma unroll
  for (int kk = 0; kk < DIMS_; kk += 4) {
    v2f a, b;
    // A 16x4 fp32: VGPR0 = K={0|2}, VGPR1 = K={1|3} (ISA 7.12.2)
    a.x = xs[mn * XS_STRIDE + kk + kb];
    a.y = xs[mn * XS_STRIDE + kk + kb + 1];
    // B 4x16 fp32: row K striped the same way, N = lane&15
    b.x = wc[(kk + kb) * 128 + n0 + mn];
    b.y = wc[(kk + kb + 1) * 128 + n0 + mn];
    c = __builtin_amdgcn_wmma_f32_16x16x4_f32(
        /*neg_a=*/false, a, /*neg_b=*/false, b,
        /*c_mod=*/(short)0, c, /*reuse_a=*/false, /*reuse_b=*/false);
  }
  // D 16x16 fp32: VGPR r -> row r (lanes 0-15) / row r+8 (lanes 16-31)
#pragma unroll
  for (int r = 0; r < 8; ++r) {
    const int m = r + 8 * (lane >> 4);
    PQ[(size_t)(rowbase + m) * 128 + n0 + mn] = c[r];
  }
}

// ---------------------------------------------------------------------------
// Kernel 4: out[b,n,:] = max_k relu(LN(P[b,idx[k],:] + Q[b,n,:] + b2))
// PQ is 16MB -> L2-resident, gathers are 256B coalesced rows.
// ---------------------------------------------------------------------------
__global__ void finalize_kernel(const float* __restrict__ PQ,
                                const int* __restrict__ idx,
                                const float* __restrict__ b2,
                                const float* __restrict__ g2,
                                const float* __restrict__ be2,
                                float* __restrict__ out) {
  const int row = blockIdx.x;              // b*N + n
  const int ch  = threadIdx.x;             // 0..63
  const int b   = row >> 12;               // N = 4096
  const int wid = ch >> 5;

  const float q   = PQ[(size_t)row * 128 + DIMS_ + ch] + b2[ch];
  const float gch = g2[ch];
  const float bch = be2[ch];

  __shared__ float red[2][2];
  __shared__ int   sj[KNB_];
  if (ch < KNB_) sj[ch] = idx[row * KNB_ + ch];
  __syncthreads();

  float best = -3.4e38f;
  for (int k = 0; k < KNB_; ++k) {
    const int j = sj[k];
    const float pre = PQ[((size_t)(b << 12) + j) * 128 + ch] + q;

    float s = pre;
#pragma unroll
    for (int o = 16; o > 0; o >>= 1) s += __shfl_xor(s, o, 32);
    if ((ch & 31) == 0) red[wid][0] = s;
    __syncthreads();
    const float m = (red[0][0] + red[1][0]) * (1.0f / DIMS_);

    const float d = pre - m;
    float s2 = d * d;
#pragma unroll
    for (int o = 16; o > 0; o >>= 1) s2 += __shfl_xor(s2, o, 32);
    if ((ch & 31) == 0) red[wid][1] = s2;
    __syncthreads();
    const float v = (red[0][1] + red[1][1]) * (1.0f / DIMS_);

    const float val = d * rsqrtf(v + EPS_) * gch + bch;
    best = fmaxf(best, fmaxf(val, 0.0f));
  }
  out[(size_t)row * DIMS_ + ch] = best;
}

// ---------------------------------------------------------------------------
extern "C" void kernel_launch(void* const* d_in, const int* in_sizes, int n_in,
                              void* d_out, int out_size, void* d_ws, size_t ws_size,
                              hipStream_t stream) {
  const float* inputs = (const float*)d_in[0];
  // d_in[1] = coordinates (unused; distance matrix is provided)
  const float* dmat   = (const float*)d_in[2];
  const float* W1     = (const float*)d_in[3];
  const float* b1     = (const float*)d_in[4];
  const float* g1     = (const float*)d_in[5];
  const float* be1    = (const float*)d_in[6];
  const float* W2     = (const float*)d_in[7];
  const float* b2     = (const float*)d_in[8];
  const float* g2     = (const float*)d_in[9];
  const float* be2    = (const float*)d_in[10];
  float* out = (float*)d_out;

  const int rows = B_ * N_;                       // 32768
  float* x  = (float*)d_ws;                       // rows*64 floats  (8 MB)
  float* PQ = x + (size_t)rows * DIMS_;           // rows*128 floats (16 MB)
  int*   id = (int*)(PQ + (size_t)rows * 2 * DIMS_); // rows*32 ints (4 MB)

  embed_kernel   <<<rows,      64,  0, stream>>>(inputs, W1, b1, g1, be1, x);
  qball_kernel   <<<rows / 8,  256, 0, stream>>>(dmat, id);
  project_kernel <<<rows / 16, 256, 0, stream>>>(x, W2, PQ);
  finalize_kernel<<<rows,      64,  0, stream>>>(PQ, id, b2, g2, be2, out);
}